// AttentionRoPE_37477884625208
// MI455X (gfx1250) — compile-verified
//
#include <hip/hip_runtime.h>
#include <math.h>

#define DIMC   1024
#define NHEADS 16
#define HDIM   64
#define HALF   32
#define BATCH  2
#define SEQ    2048
#define QKVLD  (3 * DIMC)

typedef __attribute__((ext_vector_type(2))) float v2f;
typedef __attribute__((ext_vector_type(8))) float v8f;

// ---------- WMMA helper (V_WMMA_F32_16X16X4_F32) ----------
__device__ __forceinline__ v8f wmma4(v2f a, v2f b, v8f c) {
  // 8 args: (neg_a, A, neg_b, B, c_mod, C, reuse_a, reuse_b)
  return __builtin_amdgcn_wmma_f32_16x16x4_f32(false, a, false, b, (short)0, c,
                                               false, false);
}

// A-fragment (16x4, MxK) or B^T-fragment from row-major (rows, K) storage.
// Lane L<16: (row0+L, k0..k0+1); lane>=16: (row0+L-16, k0+2..k0+3).
__device__ __forceinline__ v2f frag_nk(const float* __restrict__ base, int row0,
                                       int ld, int k0) {
  int lane = threadIdx.x & 31;
  const float* p =
      base + (size_t)(row0 + (lane & 15)) * ld + (k0 + ((lane >> 4) << 1));
  return *(const v2f*)p;
}

// B-fragment from row-major (K, N) storage (strided 2x b32 loads).
__device__ __forceinline__ v2f frag_kn(const float* __restrict__ base, int k0,
                                       int ld, int n0) {
  int lane = threadIdx.x & 31;
  int n = n0 + (lane & 15);
  int k = k0 + ((lane >> 4) << 1);
  v2f r;
  r.x = base[(size_t)k * ld + n];
  r.y = base[(size_t)(k + 1) * ld + n];
  return r;
}

// ---------- GEMM: Y[M,N] = X[M,K] @ W[N,K]^T (+ bias) ----------
// One wave computes a (BM*16) x (BN*16) block of C. BM=BN=4 gives
// 4+4 fragment loads : 16 WMMAs per K-step (~16 FLOP/byte of L2 traffic).
template <int BM, int BN>
__global__ __launch_bounds__(32) void gemm_nt_kernel(
    const float* __restrict__ X, const float* __restrict__ W,
    const float* __restrict__ bias, float* __restrict__ Y, int M, int K,
    int Ncols) {
  int c0 = blockIdx.x * (BN * 16);
  int r0 = blockIdx.y * (BM * 16);
  v8f acc[BM][BN] = {};
  int lane = threadIdx.x & 31;

  for (int k0 = 0; k0 < K; k0 += 4) {
    if ((k0 & 31) == 0) {
      // Uniform-guarded software prefetch 256B ahead (global_prefetch_b8).
      __builtin_prefetch(X + (size_t)(r0 + (lane & 15)) * K + k0 + 64, 0, 3);
      __builtin_prefetch(W + (size_t)(c0 + (lane & 15)) * K + k0 + 64, 0, 3);
    }
    v2f a[BM], b[BN];
#pragma unroll
    for (int i = 0; i < BM; ++i) a[i] = frag_nk(X, r0 + 16 * i, K, k0);
#pragma unroll
    for (int j = 0; j < BN; ++j) b[j] = frag_nk(W, c0 + 16 * j, K, k0);
#pragma unroll
    for (int i = 0; i < BM; ++i)
#pragma unroll
      for (int j = 0; j < BN; ++j) acc[i][j] = wmma4(a[i], b[j], acc[i][j]);
  }

  int col = lane & 15;
  int hl = lane >> 4;
#pragma unroll
  for (int i = 0; i < BM; ++i) {
#pragma unroll
    for (int r = 0; r < 8; ++r) {
      int row = r0 + 16 * i + r + 8 * hl;
      float* yrow = Y + (size_t)row * Ncols;
#pragma unroll
      for (int j = 0; j < BN; ++j) {
        int c = c0 + 16 * j + col;
        float bb = bias ? bias[c] : 0.f;
        yrow[c] = acc[i][j][r] + bb;
      }
    }
  }
}

// ---------- RoPE (interleaved pairs) applied in-place to q and k ----------
__global__ void rope_kernel(float* __restrict__ qkv,
                            const float* __restrict__ fcos,
                            const float* __restrict__ fsin) {
  int idx = blockIdx.x * blockDim.x + threadIdx.x;
  if (idx >= BATCH * SEQ * NHEADS * HALF) return;
  int p = idx & (HALF - 1);
  int h = (idx >> 5) & (NHEADS - 1);
  int n = (idx >> 9) & (SEQ - 1);
  int b = idx >> 20;
  float c = fcos[n * HALF + p];
  float s = fsin[n * HALF + p];
  size_t row = (size_t)(b * SEQ + n) * QKVLD;
  size_t qi = row + h * HDIM + 2 * p;
  float q0v = qkv[qi], q1v = qkv[qi + 1];
  qkv[qi] = q0v * c - q1v * s;
  qkv[qi + 1] = q0v * s + q1v * c;
  size_t ki = qi + DIMC;
  float k0v = qkv[ki], k1v = qkv[ki + 1];
  qkv[ki] = k0v * c - k1v * s;
  qkv[ki + 1] = k0v * s + k1v * c;
}

// ---------- Flash-style attention: one wave per (b, h, 16-query tile) ------
__global__ __launch_bounds__(32) void attn_kernel(const float* __restrict__ qkv,
                                                  float* __restrict__ out) {
  int qt = blockIdx.x, h = blockIdx.y, b = blockIdx.z;
  int q0 = qt * 16;
  const float* qb = qkv + (size_t)b * SEQ * QKVLD + h * HDIM;
  const float* kb = qb + DIMC;
  const float* vb = qb + 2 * DIMC;
  __shared__ __align__(16) float ps[16][18];  // even stride -> 8B-aligned v2f
  int lane = threadIdx.x & 31;
  int col = lane & 15;
  int hl = lane >> 4;

  // Q tile held in registers as 16 A-fragments (K = 64, steps of 4)
  v2f qf[16];
#pragma unroll
  for (int kk = 0; kk < 16; ++kk) qf[kk] = frag_nk(qb, q0, QKVLD, kk * 4);

  v8f o0 = {}, o1 = {}, o2 = {}, o3 = {};
  float mrow[8], lrow[8];
#pragma unroll
  for (int r = 0; r < 8; ++r) {
    mrow[r] = -3.0e38f;
    lrow[r] = 0.f;
  }
  const float scale = 0.125f;  // HEAD_DIM^-0.5

  for (int kt = 0; kt < SEQ / 16; ++kt) {
    int j0 = kt * 16;
    // S = Q @ K^T (16x16 tile)
    v8f s = {};
#pragma unroll
    for (int kk = 0; kk < 16; ++kk) {
      v2f kf = frag_nk(kb, j0, QKVLD, kk * 4);
      s = wmma4(qf[kk], kf, s);
    }
    // Online softmax: rows of the C-layout live within 16-lane halves.
    float p[8];
#pragma unroll
    for (int r = 0; r < 8; ++r) {
      float v = s[r] * scale;
      float mx = v;
      mx = fmaxf(mx, __shfl_xor(mx, 1, 32));
      mx = fmaxf(mx, __shfl_xor(mx, 2, 32));
      mx = fmaxf(mx, __shfl_xor(mx, 4, 32));
      mx = fmaxf(mx, __shfl_xor(mx, 8, 32));
      float mnew = fmaxf(mrow[r], mx);
      float alpha = __expf(mrow[r] - mnew);
      float pv = __expf(v - mnew);
      float rs = pv;
      rs += __shfl_xor(rs, 1, 32);
      rs += __shfl_xor(rs, 2, 32);
      rs += __shfl_xor(rs, 4, 32);
      rs += __shfl_xor(rs, 8, 32);
      lrow[r] = lrow[r] * alpha + rs;
      mrow[r] = mnew;
      p[r] = pv;
      o0[r] *= alpha;
      o1[r] *= alpha;
      o2[r] *= alpha;
      o3[r] *= alpha;
    }
    // Re-shape P: C-layout -> A-layout via LDS
#pragma unroll
    for (int r = 0; r < 8; ++r) ps[r + 8 * hl][col] = p[r];
    __syncthreads();
    v2f pf[4];
#pragma unroll
    for (int kk = 0; kk < 4; ++kk) {
      const float* pp = &ps[col][kk * 4 + hl * 2];
      pf[kk] = *(const v2f*)pp;
    }
    __syncthreads();
    // O += P @ V  (V chunk is row-major (K=16 keys, N=64 dims))
    const float* vbase = vb + (size_t)j0 * QKVLD;
#pragma unroll
    for (int kk = 0; kk < 4; ++kk) {
      o0 = wmma4(pf[kk], frag_kn(vbase, kk * 4, QKVLD, 0), o0);
      o1 = wmma4(pf[kk], frag_kn(vbase, kk * 4, QKVLD, 16), o1);
      o2 = wmma4(pf[kk], frag_kn(vbase, kk * 4, QKVLD, 32), o2);
      o3 = wmma4(pf[kk], frag_kn(vbase, kk * 4, QKVLD, 48), o3);
    }
  }
  // Normalize and write (b, n, h, d) == row-major (4096, 1024)
#pragma unroll
  for (int r = 0; r < 8; ++r) {
    float inv = 1.0f / lrow[r];
    int n = q0 + r + 8 * hl;
    size_t base = (size_t)(b * SEQ + n) * DIMC + h * HDIM + col;
    out[base + 0] = o0[r] * inv;
    out[base + 16] = o1[r] * inv;
    out[base + 32] = o2[r] * inv;
    out[base + 48] = o3[r] * inv;
  }
}

extern "C" void kernel_launch(void* const* d_in, const int* in_sizes, int n_in,
                              void* d_out, int out_size, void* d_ws,
                              size_t ws_size, hipStream_t stream) {
  const float* x = (const float*)d_in[0];
  const float* fcos = (const float*)d_in[1];
  const float* fsin = (const float*)d_in[2];
  const float* Wqkv = (const float*)d_in[3];
  const float* Wproj = (const float*)d_in[4];
  const float* bproj = (const float*)d_in[5];
  float* out = (float*)d_out;

  float* qkv = (float*)d_ws;                        // 4096*3072 floats (48 MB)
  float* attn = qkv + (size_t)BATCH * SEQ * QKVLD;  // 4096*1024 floats (16 MB)

  const int M = BATCH * SEQ;

  // 1) QKV projection: qkv = x @ Wqkv^T  (64x64 tile/wave, high reuse)
  gemm_nt_kernel<4, 4><<<dim3(QKVLD / 64, M / 64), 32, 0, stream>>>(
      x, Wqkv, nullptr, qkv, M, DIMC, QKVLD);

  // 2) RoPE on q and k (in place)
  int total = BATCH * SEQ * NHEADS * HALF;
  rope_kernel<<<(total + 255) / 256, 256, 0, stream>>>(qkv, fcos, fsin);

  // 3) Flash attention -> attn (b, n, h, d)
  attn_kernel<<<dim3(SEQ / 16, NHEADS, BATCH), 32, 0, stream>>>(qkv, attn);

  // 4) Output projection with bias (32x32 tile/wave for a wider grid)
  gemm_nt_kernel<2, 2><<<dim3(DIMC / 32, M / 32), 32, 0, stream>>>(
      attn, Wproj, bproj, out, M, DIMC, DIMC);
}